// InvariantPriorEncoder_24592982737421
// MI455X (gfx1250) — compile-verified
//
#include <hip/hip_runtime.h>
#include <hip/hip_bf16.h>

typedef float v2f __attribute__((ext_vector_type(2)));
typedef float v8f __attribute__((ext_vector_type(8)));

#define Q 16
#define D 32
#define H1C 128
#define H2C 256
#define NBLK 1024
#define NREP 4

#define SX_STRIDE 36    // 32 + 4 pad: 36*row mod 64 distinct for 16 rows
#define SH_STRIDE 130   // 128 + 2 pad
#define SAB_STRIDE 260  // 256 + 4 pad (float4-aligned rows)
#define SL_STRIDE 17

#define WMMA_F32X4(c, a, b) \
  c = __builtin_amdgcn_wmma_f32_16x16x4_f32(false, a, false, b, (short)0, c, false, false)

__global__ __launch_bounds__(128)
void ipe_fused_kernel(const float* __restrict__ X,
                      const int* __restrict__ t0s,
                      const float* __restrict__ eps_z,
                      const float* __restrict__ mW1, const float* __restrict__ mb1,
                      const float* __restrict__ mW2, const float* __restrict__ mb2,
                      const float* __restrict__ cW1, const float* __restrict__ cb1,
                      const float* __restrict__ cW2, const float* __restrict__ cb2,
                      float* __restrict__ out)
{
  __shared__ float sX[16 * SX_STRIDE];     // window 16x32
  __shared__ float sH[16 * SH_STRIDE];     // mu hidden 16x128
  __shared__ float sA[16 * SAB_STRIDE];    // Xwin @ cW1_top + cb1
  __shared__ float sB[16 * SAB_STRIDE];    // Xwin @ cW1_bot
  __shared__ float sW2T[16 * SAB_STRIDE];  // cW2 transposed: [j][k]
  __shared__ float sL[16 * SL_STRIDE];     // tril L
  __shared__ float sCov[16 * SL_STRIDE];
  __shared__ float sMu[16];

  const int tid  = threadIdx.x;
  const int lane = tid & 31;
  const int w    = tid >> 5;      // wave id 0..3
  const int rowm = lane & 15;     // M-row (A,C/D) or N-col (B,C/D) within tile
  const int hi   = lane >> 4;     // half-wave selector
  const int n    = blockIdx.x;

  // Preload cW2 transposed into LDS (coalesced global read)
  for (int idx = tid; idx < H2C * Q; idx += 128) {
    int k = idx >> 4, j = idx & 15;
    sW2T[j * SAB_STRIDE + k] = cW2[idx];
  }
  if (tid < 16) sMu[tid] = 0.f;

  v8f covC = {};  // cov accumulator over replicas (wave 0)

  for (int r = 0; r < NREP; ++r) {
    const int s  = r * NBLK + n;
    const int t0 = t0s[s];
    const float* xw = X + ((size_t)n * 64 + t0) * D;

    __syncthreads();  // protect sX/sL reuse (and sW2T/sMu init on first pass)

    // Load 16x32 window: 512 consecutive floats, one float4 per thread
    {
      int e  = tid * 4;
      int rr = e >> 5, cc = e & 31;
      const float4 v = *(const float4*)(xw + e);
      *(float4*)(&sX[rr * SX_STRIDE + cc]) = v;
    }
    __syncthreads();

    // ---- Stage A1: H = relu(Xwin @ mW1 + mb1), 8 N-tiles, 2 per wave ----
    for (int t = 0; t < 2; ++t) {
      const int nt = w * 2 + t;
      v8f c = {};
      #pragma unroll
      for (int kk = 0; kk < 8; ++kk) {
        const int k0 = kk * 4 + 2 * hi;
        v2f a; a.x = sX[rowm * SX_STRIDE + k0];  a.y = sX[rowm * SX_STRIDE + k0 + 1];
        v2f b; b.x = mW1[k0 * H1C + nt * 16 + rowm];
               b.y = mW1[(k0 + 1) * H1C + nt * 16 + rowm];
        WMMA_F32X4(c, a, b);
      }
      const float bias = mb1[nt * 16 + rowm];
      #pragma unroll
      for (int q = 0; q < 8; ++q) {
        float v = c[q] + bias;
        sH[(q + 8 * hi) * SH_STRIDE + nt * 16 + rowm] = v > 0.f ? v : 0.f;
      }
    }

    // ---- Stage B1: A = Xwin@cW1_top + cb1, B = Xwin@cW1_bot (32 tiles, 8/wave) ----
    for (int t = 0; t < 8; ++t) {
      const int tt   = w * 8 + t;
      const int isB  = tt >= 16;
      const int nt   = tt & 15;
      const int koff = isB ? 32 : 0;
      v8f c = {};
      #pragma unroll
      for (int kk = 0; kk < 8; ++kk) {
        const int k0 = kk * 4 + 2 * hi;
        v2f a; a.x = sX[rowm * SX_STRIDE + k0];  a.y = sX[rowm * SX_STRIDE + k0 + 1];
        v2f b; b.x = cW1[(koff + k0) * H2C + nt * 16 + rowm];
               b.y = cW1[(koff + k0 + 1) * H2C + nt * 16 + rowm];
        WMMA_F32X4(c, a, b);
      }
      const float bias = isB ? 0.f : cb1[nt * 16 + rowm];
      float* dst = isB ? sB : sA;
      #pragma unroll
      for (int q = 0; q < 8; ++q)
        dst[(q + 8 * hi) * SAB_STRIDE + nt * 16 + rowm] = c[q] + bias;
    }
    __syncthreads();

    // ---- Stage A2 (wave 0): M = H @ mW2, mu_s = rowmean(M) + mb2 ----
    if (w == 0) {
      v8f c = {};
      for (int kk = 0; kk < 32; ++kk) {
        const int k0 = kk * 4 + 2 * hi;
        v2f a; a.x = sH[rowm * SH_STRIDE + k0];  a.y = sH[rowm * SH_STRIDE + k0 + 1];
        v2f b; b.x = mW2[k0 * Q + rowm];  b.y = mW2[(k0 + 1) * Q + rowm];
        WMMA_F32X4(c, a, b);
      }
      float p = 0.f;
      #pragma unroll
      for (int q = 0; q < 8; ++q) p += c[q];      // sum of 8 rows, this column
      p += __shfl_xor(p, 16, 32);                 // + other 8 rows (same column)
      if (hi == 0) sMu[rowm] += p * (1.f / 16.f) + mb2[rowm];
    }

    // ---- Stage B2: pairs. L[i][j] = relu(A[i]+B[j]) . cW2[:,j] + cb2[j], i>=j ----
    for (int pp = 0; pp < 64; ++pp) {
      const int p = w * 64 + pp;
      const int i = p >> 4, j = p & 15;
      float lij = 0.f;
      if (i >= j) {                                // wave-uniform branch
        float acc = 0.f;
        #pragma unroll
        for (int e = 0; e < 8; ++e) {
          const int k = e * 32 + lane;             // conflict-free LDS access
          float h = sA[i * SAB_STRIDE + k] + sB[j * SAB_STRIDE + k];
          h = h > 0.f ? h : 0.f;
          acc += h * sW2T[j * SAB_STRIDE + k];
        }
        #pragma unroll
        for (int m = 16; m > 0; m >>= 1) acc += __shfl_xor(acc, m, 32);
        lij = acc + cb2[j];
      }
      if (lane == 0) sL[i * SL_STRIDE + j] = lij;
    }
    __syncthreads();

    // ---- Stage B3 (wave 0): covC += L @ L^T  (B operand == A operand) ----
    if (w == 0) {
      #pragma unroll
      for (int kk = 0; kk < 4; ++kk) {
        const int k0 = kk * 4 + 2 * hi;
        v2f a; a.x = sL[rowm * SL_STRIDE + k0];  a.y = sL[rowm * SL_STRIDE + k0 + 1];
        WMMA_F32X4(covC, a, a);   // B[k][n] = L[n][k] == A layout with n=rowm
      }
    }
  }

  __syncthreads();
  // Average cov, add eps*I, spill to LDS
  if (w == 0) {
    #pragma unroll
    for (int q = 0; q < 8; ++q) {
      const int i = q + 8 * hi;
      float v = covC[q] * 0.25f;
      if (i == rowm) v += 1e-6f;
      sCov[i * SL_STRIDE + rowm] = v;
    }
  }
  __syncthreads();

  // ---- Cholesky 16x16 (cooperative, lower triangle in place) ----
  for (int k = 0; k < 16; ++k) {
    if (tid == 0) sCov[k * SL_STRIDE + k] = sqrtf(sCov[k * SL_STRIDE + k]);
    __syncthreads();
    if (tid > k && tid < 16) sCov[tid * SL_STRIDE + k] /= sCov[k * SL_STRIDE + k];
    __syncthreads();
    for (int e = tid; e < 256; e += 128) {
      const int i = e >> 4, j = e & 15;
      if (i > k && j > k && j <= i)
        sCov[i * SL_STRIDE + j] -= sCov[i * SL_STRIDE + k] * sCov[j * SL_STRIDE + k];
    }
    __syncthreads();
  }

  // ---- z = mu + Lc @ eps_z ----
  if (tid < 16) {
    const int i = tid;
    float z = sMu[i] * 0.25f;
    for (int j = 0; j <= i; ++j)
      z += sCov[i * SL_STRIDE + j] * eps_z[n * 16 + j];
    out[n * 16 + i] = z;
  }
}

extern "C" void kernel_launch(void* const* d_in, const int* in_sizes, int n_in,
                              void* d_out, int out_size, void* d_ws, size_t ws_size,
                              hipStream_t stream) {
  const float* X     = (const float*)d_in[0];
  const int*   t0s   = (const int*)  d_in[1];
  const float* eps_z = (const float*)d_in[2];
  const float* mW1   = (const float*)d_in[3];
  const float* mb1   = (const float*)d_in[4];
  const float* mW2   = (const float*)d_in[5];
  const float* mb2   = (const float*)d_in[6];
  const float* cW1   = (const float*)d_in[7];
  const float* cb1   = (const float*)d_in[8];
  const float* cW2   = (const float*)d_in[9];
  const float* cb2   = (const float*)d_in[10];
  float* out = (float*)d_out;

  ipe_fused_kernel<<<NBLK, 128, 0, stream>>>(X, t0s, eps_z, mW1, mb1, mW2, mb2,
                                             cW1, cb1, cW2, cb2, out);
}